// MixedFeatureEmbedder_53008486367532
// MI455X (gfx1250) — compile-verified
//
#include <hip/hip_runtime.h>

typedef float v4f __attribute__((ext_vector_type(4)));

constexpr int TOKEN_DIM = 128;
constexpr int T4       = TOKEN_DIM / 4;   // 32 float4 per token row == one wave32
constexpr int NFEAT    = 64;
constexpr int NNUM     = 32;              // numeric (== categorical) feature count
constexpr int CARD     = 100;
constexpr int TILE_B   = 8;               // batch rows per block
constexpr int ROWS     = TILE_B * NFEAT;  // 512 (b,f) token-rows per block
constexpr int THREADS  = 256;             // 8 wave32 per block
constexpr int WAVES    = THREADS / 32;

__global__ __launch_bounds__(THREADS) void mixed_embed_kernel(
    const float* __restrict__ x,
    const float* __restrict__ Wn,
    const float* __restrict__ bn,
    const float* __restrict__ emb,
    float* __restrict__ out)
{
    __shared__ v4f      sW[NNUM * T4];   // 16 KB  W_num
    __shared__ v4f      sB[NNUM * T4];   // 16 KB  b_num
    __shared__ unsigned sMeta[ROWS];     //  2 KB  per-row scalar / gather offset

    const int tid = threadIdx.x;
    const int b0  = blockIdx.x * TILE_B;

    // ---- CDNA5 async global->LDS stage of W_num + b_num (32 KB total) ----
    // 2048 x 16B transfers; each of 256 lanes issues 4+4 async b128 copies.
    {
        const unsigned ldsW = (unsigned)(size_t)&sW[0];
        const unsigned ldsB = (unsigned)(size_t)&sB[0];
        #pragma unroll
        for (int i = 0; i < 4; ++i) {
            const unsigned e  = (unsigned)(tid + i * THREADS) * 16u; // byte offset
            const unsigned lw = ldsW + e;
            const unsigned lb = ldsB + e;
            asm volatile("global_load_async_to_lds_b128 %0, %1, %2"
                         :: "v"(lw), "v"(e), "s"(Wn) : "memory");
            asm volatile("global_load_async_to_lds_b128 %0, %1, %2"
                         :: "v"(lb), "v"(e), "s"(bn) : "memory");
        }
    }

    // ---- Per-(b,f) metadata: numeric -> scalar bits; categorical -> element offset
    for (int p = tid; p < ROWS; p += THREADS) {
        const int   bl = p >> 6;            // 0..TILE_B-1
        const int   f  = p & (NFEAT - 1);
        const float xv = x[(size_t)(b0 + bl) * NFEAT + f];
        unsigned m;
        if ((f & 1) == 0) {
            m = __float_as_uint(xv);        // broadcast scalar for x*W+b
        } else {
            float v = (xv != xv) ? 0.0f : xv;            // nan_to_num
            float r = rintf(v);                          // round half-to-even
            r = fminf(fmaxf(r, 0.0f), (float)(CARD - 1));// clip [0, 99]
            const int idx = (int)r;
            const int j   = f >> 1;
            m = (unsigned)((j * CARD + idx) * TOKEN_DIM);// float-element offset
        }
        sMeta[p] = m;
    }

    asm volatile("s_wait_asynccnt 0x0" ::: "memory"); // async copies landed in LDS
    __syncthreads();

    // ---- Streaming phase: one wave per (b,f) row, lane = float4 in token ----
    const int  wave = tid >> 5;
    const int  lane = tid & 31;
    const v4f* __restrict__ embv = (const v4f*)emb;
    v4f*       __restrict__ outv = (v4f*)out;
    const size_t rowBase = (size_t)b0 * NFEAT * T4;

    #pragma unroll 4
    for (int i = 0; i < ROWS / WAVES; ++i) {
        const int      p = i * WAVES + wave;      // waves write adjacent rows
        const int      f = p & (NFEAT - 1);
        const unsigned m = sMeta[p];
        v4f v;
        if ((f & 1) == 0) {
            const int   j  = f >> 1;
            const float s  = __uint_as_float(m);
            const v4f   w4 = sW[j * T4 + lane];
            const v4f   c4 = sB[j * T4 + lane];
            v.x = fmaf(s, w4.x, c4.x);
            v.y = fmaf(s, w4.y, c4.y);
            v.z = fmaf(s, w4.z, c4.z);
            v.w = fmaf(s, w4.w, c4.w);
        } else {
            v = embv[(size_t)(m >> 2) + lane];    // 512B coalesced gather (L2-hot)
        }

        // Prefetch next categorical row into WGP$ (global_prefetch_b8).
        const int pn = p + WAVES;
        if (pn < ROWS && (pn & 1)) {
            __builtin_prefetch(&embv[(size_t)(sMeta[pn] >> 2) + lane], 0, 0);
        }

        // 256MB output stream > 192MB L2 -> non-temporal 128-bit store.
        __builtin_nontemporal_store(v, &outv[rowBase + (size_t)p * T4 + lane]);
    }
}

extern "C" void kernel_launch(void* const* d_in, const int* in_sizes, int n_in,
                              void* d_out, int out_size, void* d_ws, size_t ws_size,
                              hipStream_t stream)
{
    const float* x   = (const float*)d_in[0];
    const float* Wn  = (const float*)d_in[1];
    const float* bn  = (const float*)d_in[2];
    const float* emb = (const float*)d_in[3];
    float*       out = (float*)d_out;

    const int batch  = in_sizes[0] / NFEAT;     // 8192
    const int blocks = batch / TILE_B;          // 1024

    mixed_embed_kernel<<<blocks, THREADS, 0, stream>>>(x, Wn, bn, emb, out);
}